// DTSHLoss_48189533061230
// MI455X (gfx1250) — compile-verified
//
#include <hip/hip_runtime.h>

#define B_    384
#define BITS_ 128
#define NCLS_ 80
#define NT_   (B_ / 16)   // 24 tiles per dimension

typedef __attribute__((ext_vector_type(2))) float v2f;
typedef __attribute__((ext_vector_type(8))) float v8f;

// ---------------------------------------------------------------------------
// Zero the scalar accumulators in workspace (must run every call: the harness
// graph-replays kernel_launch and we accumulate with atomics).
// ---------------------------------------------------------------------------
__global__ void init_ws_kernel(float* ws) {
    if (threadIdx.x < 4) ws[threadIdx.x] = 0.0f;
}

// ---------------------------------------------------------------------------
// Pack y[B, 80] (int 0/1) into 3 u32 bitmask words per row.
// Similarity s[i][j] = (y_i . y_j > 0) == any shared bit -> AND + nonzero.
// ---------------------------------------------------------------------------
__global__ void pack_y_kernel(const int* __restrict__ y, unsigned* __restrict__ ybits) {
    int r = blockIdx.x * blockDim.x + threadIdx.x;
    if (r >= B_) return;
    const int* yr = y + r * NCLS_;
    unsigned w0 = 0, w1 = 0, w2 = 0;
    #pragma unroll
    for (int c = 0; c < NCLS_; ++c) {
        unsigned bit = (yr[c] != 0) ? 1u : 0u;
        if (c < 32)      w0 |= bit << c;
        else if (c < 64) w1 |= bit << (c - 32);
        else             w2 |= bit << (c - 64);
    }
    ybits[r * 3 + 0] = w0;
    ybits[r * 3 + 1] = w1;
    ybits[r * 3 + 2] = w2;
}

// ---------------------------------------------------------------------------
// ip = u @ u^T via V_WMMA_F32_16X16X4_F32.
// One wave32 -> one 16x16 output tile; K=128 in steps of 4 (32 WMMAs/wave).
// A layout (16x4 f32): lane[0:15] M=lane, VGPR0/1 = K0/K1; lane[16:31] K2/K3.
// B = u^T, so the B operand is the identical per-lane float2 load with the
// N-tile row index. C/D: VGPR r, half h -> row 16*tm + 8*h + r, col lane&15.
// ---------------------------------------------------------------------------
__global__ __launch_bounds__(256) void ip_wmma_kernel(const float* __restrict__ u,
                                                      float* __restrict__ ip) {
    const int wave = threadIdx.x >> 5;
    const int lane = threadIdx.x & 31;
    const int tile = blockIdx.x * 8 + wave;      // 72 blocks * 8 waves = 576 tiles
    const int tm = tile / NT_;
    const int tn = tile % NT_;
    const int half = lane >> 4;                  // 0: K={0,1}, 1: K={2,3}
    const int l    = lane & 15;

    const float* arow = u + (tm * 16 + l) * BITS_;
    const float* brow = u + (tn * 16 + l) * BITS_;

    __builtin_prefetch(arow, 0, 0);              // global_prefetch_b8
    __builtin_prefetch(brow, 0, 0);

    v8f c = {};
    #pragma unroll 8
    for (int k0 = 0; k0 < BITS_; k0 += 4) {
        v2f a = *(const v2f*)(arow + k0 + 2 * half);
        v2f b = *(const v2f*)(brow + k0 + 2 * half);
        // (neg_a, A, neg_b, B, c_mod, C, reuse_a, reuse_b)
        c = __builtin_amdgcn_wmma_f32_16x16x4_f32(false, a, false, b,
                                                  (short)0, c, false, false);
    }

    #pragma unroll
    for (int r = 0; r < 8; ++r)
        ip[(tm * 16 + 8 * half + r) * B_ + tn * 16 + l] = c[r];
}

// ---------------------------------------------------------------------------
// Per-row loss: one block (8 wave32s) per row r.
//   row_loss[r] = sum_{i in Pos(r), j in Neg(r)} softplus(-(ip[r,i]-ip[r,j]-0.5))
//                 / (n_pos * n_neg)
// Instead of scanning all B^2 pairs with a mask branch, compact the Pos and
// Neg index lists in LDS (deterministic prefix-sum over 12 mask words) and
// iterate only the npos*nneg cross product, parallelizing the 256 threads
// over the LONGER list so lanes stay busy in both skew regimes.
// ---------------------------------------------------------------------------
__global__ __launch_bounds__(256) void row_loss_kernel(const float* __restrict__ ip,
                                                       const unsigned* __restrict__ ybits,
                                                       float* ws) {
    const int r   = blockIdx.x;
    const int tid = threadIdx.x;

    __shared__ float    arow[B_];
    __shared__ unsigned posm[B_ / 32];
    __shared__ short    plist[B_];
    __shared__ short    nlist[B_];
    __shared__ int      pbase[B_ / 32];
    __shared__ int      nbase[B_ / 32];
    __shared__ int      npos_s;
    __shared__ float    wred[8];

    const unsigned yr0 = ybits[r * 3 + 0];
    const unsigned yr1 = ybits[r * 3 + 1];
    const unsigned yr2 = ybits[r * 3 + 2];

    for (int i = tid; i < B_; i += 256) arow[i] = ip[r * B_ + i];

    if (tid < B_ / 32) {
        unsigned w = 0;
        const int base = tid * 32;
        #pragma unroll
        for (int k = 0; k < 32; ++k) {
            const int i = base + k;
            const unsigned sim = (ybits[i * 3 + 0] & yr0) |
                                 (ybits[i * 3 + 1] & yr1) |
                                 (ybits[i * 3 + 2] & yr2);
            w |= (unsigned)(sim != 0u) << k;
        }
        posm[tid] = w;
    }
    __syncthreads();

    if (tid == 0) {   // exclusive prefix sums over 12 words (deterministic)
        int pc = 0, nc = 0;
        #pragma unroll
        for (int w = 0; w < B_ / 32; ++w) {
            pbase[w] = pc; nbase[w] = nc;
            const int c = __popc(posm[w]);
            pc += c; nc += 32 - c;
        }
        npos_s = pc;
    }
    __syncthreads();

    if (tid < B_ / 32) {   // scatter compacted indices
        const unsigned w = posm[tid];
        int pi = pbase[tid], ni = nbase[tid];
        #pragma unroll
        for (int k = 0; k < 32; ++k) {
            const short idx = (short)(tid * 32 + k);
            if ((w >> k) & 1u) plist[pi++] = idx;
            else               nlist[ni++] = idx;
        }
    }
    __syncthreads();

    const int npos = npos_s;
    const int nneg = B_ - npos;

    float acc = 0.0f;
    if (npos > 0 && nneg > 0) {
        if (nneg >= npos) {
            // serialize over pos (short list), parallelize over neg
            for (int ii = 0; ii < npos; ++ii) {
                const float ai = arow[plist[ii]];
                for (int jj = tid; jj < nneg; jj += 256) {
                    float t = ai - arow[nlist[jj]] - 0.5f;
                    t = fminf(fmaxf(t, -100.0f), 50.0f);
                    const float x = -t;   // softplus(x), numerically stable
                    acc += fmaxf(x, 0.0f) + __logf(1.0f + __expf(-fabsf(x)));
                }
            }
        } else {
            // serialize over neg (short list), parallelize over pos
            for (int jj = 0; jj < nneg; ++jj) {
                const float aj = arow[nlist[jj]];
                for (int ii = tid; ii < npos; ii += 256) {
                    float t = arow[plist[ii]] - aj - 0.5f;
                    t = fminf(fmaxf(t, -100.0f), 50.0f);
                    const float x = -t;
                    acc += fmaxf(x, 0.0f) + __logf(1.0f + __expf(-fabsf(x)));
                }
            }
        }
    }

    // wave32 reduction, then cross-wave via LDS
    #pragma unroll
    for (int off = 16; off > 0; off >>= 1) acc += __shfl_down(acc, off, 32);
    if ((tid & 31) == 0) wred[tid >> 5] = acc;
    __syncthreads();

    if (tid == 0) {
        float tot = 0.0f;
        #pragma unroll
        for (int w = 0; w < 8; ++w) tot += wred[w];
        const float pc = (float)npos * (float)nneg;
        if (pc > 0.0f) {
            atomicAdd(ws + 0, tot / pc);   // sum of valid row losses
            atomicAdd(ws + 1, 1.0f);       // count of valid rows
        }
    }
}

// ---------------------------------------------------------------------------
// loss2 partial: sum (u - sign(u))^2 over all 384*128 elements.
// ---------------------------------------------------------------------------
__global__ __launch_bounds__(256) void reg_loss_kernel(const float* __restrict__ u,
                                                       float* ws) {
    const int stride = gridDim.x * blockDim.x;
    float acc = 0.0f;
    for (int i = blockIdx.x * blockDim.x + threadIdx.x; i < B_ * BITS_; i += stride) {
        const float v = u[i];
        const float s = (v > 0.0f) ? 1.0f : ((v < 0.0f) ? -1.0f : 0.0f);
        const float d = v - s;
        acc += d * d;
    }
    #pragma unroll
    for (int off = 16; off > 0; off >>= 1) acc += __shfl_down(acc, off, 32);
    if ((threadIdx.x & 31) == 0) atomicAdd(ws + 2, acc);
}

// ---------------------------------------------------------------------------
// Combine: loss1 + loss2.
// ---------------------------------------------------------------------------
__global__ void finalize_kernel(const float* ws, float* out) {
    const float count = ws[1];
    const float loss1 = (count > 0.0f) ? ws[0] / fmaxf(count, 1.0f) : 0.0f;
    const float loss2 = 0.1f * ws[2] / (float)(B_ * BITS_);
    out[0] = loss1 + loss2;
}

// ---------------------------------------------------------------------------
extern "C" void kernel_launch(void* const* d_in, const int* in_sizes, int n_in,
                              void* d_out, int out_size, void* d_ws, size_t ws_size,
                              hipStream_t stream) {
    const float* u = (const float*)d_in[0];   // [384, 128] f32
    const int*   y = (const int*)d_in[1];     // [384, 80]  i32

    float*    ws    = (float*)d_ws;           // ws[0]=sum row_loss, ws[1]=count, ws[2]=reg
    float*    ip    = ws + 16;                // [384*384] f32, 64B-aligned offset
    unsigned* ybits = (unsigned*)(ip + B_ * B_);  // [384*3] u32

    init_ws_kernel<<<1, 32, 0, stream>>>(ws);
    pack_y_kernel<<<2, 192, 0, stream>>>(y, ybits);
    ip_wmma_kernel<<<(NT_ * NT_) / 8, 256, 0, stream>>>(u, ip);
    row_loss_kernel<<<B_, 256, 0, stream>>>(ip, ybits, ws);
    reg_loss_kernel<<<48, 256, 0, stream>>>(u, ws);
    finalize_kernel<<<1, 1, 0, stream>>>(ws, (float*)d_out);
}